// D_loss_67551245631962
// MI455X (gfx1250) — compile-verified
//
#include <hip/hip_runtime.h>

#define B_DIM 2048
#define T_DIM 8192
#define TPB   256
#define NWAVE (TPB / 32)

typedef __attribute__((ext_vector_type(2))) float v2f;
typedef __attribute__((ext_vector_type(8))) float v8f;

// Exact f32 wave32 reduction through the matrix pipe.
// A-matrix 16x4 f32 layout: VGPR0 holds K=0 (lanes 0-15) and K=2 (lanes 16-31),
// VGPR1 holds K=1 / K=3.  With a = {acc, 0} per lane:
//   A[m][0] = acc[lane m], A[m][2] = acc[lane 16+m], A[m][1] = A[m][3] = 0.
// B = all-ones 4x16  =>  D[m][n] = acc[m] + acc[m+16]  for every n.
// Per-lane sum of the 8 D VGPRs: lane n (<16) = sum over m=0..7,
// lane 16+n = sum over m=8..15.  One xor-16 shuffle completes the wave sum.
__device__ __forceinline__ float wave_reduce_wmma(float acc) {
    v2f a; a[0] = acc;  a[1] = 0.0f;
    v2f b; b[0] = 1.0f; b[1] = 1.0f;
    v8f c = {};
    v8f d = __builtin_amdgcn_wmma_f32_16x16x4_f32(
        /*neg_a=*/false, a, /*neg_b=*/false, b,
        /*c_mod=*/(short)0, c, /*reuse_a=*/false, /*reuse_b=*/false);
    float s = ((d[0] + d[1]) + (d[2] + d[3])) + ((d[4] + d[5]) + (d[6] + d[7]));
    s += __shfl_xor(s, 16, 32);
    return s;  // every lane now holds the full 32-lane sum
}

// Kernel 1: d_out[0] = 0.2 * sum((target_class - pred_class)^2); also clears poison.
__global__ __launch_bounds__(TPB) void class_init_kernel(
    const float* __restrict__ tc, const float* __restrict__ pc,
    float* __restrict__ out, int n) {
    __shared__ float sred[NWAVE];
    float acc = 0.0f;
    for (int i = threadIdx.x; i < n; i += TPB) {   // n = 6144, uniform trip count
        float d = tc[i] - pc[i];
        acc = fmaf(d, d, acc);
    }
    float ws = wave_reduce_wmma(acc);
    if ((threadIdx.x & 31) == 0) sred[threadIdx.x >> 5] = ws;
    __syncthreads();
    if (threadIdx.x == 0) {
        float tot = 0.0f;
#pragma unroll
        for (int w = 0; w < NWAVE; ++w) tot += sred[w];
        out[0] = 0.2f * tot;  // plain store: initializes the accumulator
    }
}

// Kernel 2: one block per row.  Stage diff (+zero halo) in LDS, 5-tap WMA,
// square-accumulate, WMMA wave reduction, one float atomic per block.
__global__ __launch_bounds__(TPB) void angle_kernel(
    const float* __restrict__ ta, const float* __restrict__ pa,
    float* __restrict__ out) {
    __shared__ float sdiff[T_DIM + 4];   // sdiff[k] == zero-padded diff dp[k]
    __shared__ float sred[NWAVE];

    const int tid  = threadIdx.x;
    const size_t base = (size_t)blockIdx.x * T_DIM;
    const float4* __restrict__ ta4 = (const float4*)(ta + base);
    const float4* __restrict__ pa4 = (const float4*)(pa + base);

    if (tid < 2) {                       // zero halo, 2 each side
        sdiff[tid] = 0.0f;
        sdiff[T_DIM + 2 + tid] = 0.0f;
    }

#pragma unroll
    for (int c = 0; c < T_DIM / (TPB * 4); ++c) {   // 8 coalesced b128 loads each
        int i = tid + c * TPB;                       // float4 index
        float4 t4 = ta4[i];
        float4 p4 = pa4[i];
        int o = 2 + i * 4;
        sdiff[o + 0] = t4.x - p4.x;
        sdiff[o + 1] = t4.y - p4.y;
        sdiff[o + 2] = t4.z - p4.z;
        sdiff[o + 3] = t4.w - p4.w;
    }
    __syncthreads();

    float acc = 0.0f;
#pragma unroll 4
    for (int c = 0; c < T_DIM / TPB; ++c) {          // 32 outputs per thread
        int t = tid + c * TPB;                       // conflict-free: lanes adjacent
        float wma = 0.05f * sdiff[t]
                  + 0.10f * sdiff[t + 1]
                  + 0.70f * sdiff[t + 2]
                  + 0.10f * sdiff[t + 3]
                  + 0.05f * sdiff[t + 4];
        acc = fmaf(wma, wma, acc);
    }

    float ws = wave_reduce_wmma(acc);                // uniform flow: EXEC all ones
    if ((tid & 31) == 0) sred[tid >> 5] = ws;
    __syncthreads();
    if (tid == 0) {
        float tot = 0.0f;
#pragma unroll
        for (int w = 0; w < NWAVE; ++w) tot += sred[w];
        atomicAdd(out, 0.8f * tot);
    }
}

extern "C" void kernel_launch(void* const* d_in, const int* in_sizes, int n_in,
                              void* d_out, int out_size, void* d_ws, size_t ws_size,
                              hipStream_t stream) {
    const float* target_angle = (const float*)d_in[0];  // [B, T]
    const float* target_class = (const float*)d_in[1];  // [B, 3]
    const float* pred_angle   = (const float*)d_in[2];  // [B, T]
    const float* pred_class   = (const float*)d_in[3];  // [B, 3]
    float* out = (float*)d_out;

    // Stream order: init store completes before any angle-kernel atomics.
    class_init_kernel<<<1, TPB, 0, stream>>>(target_class, pred_class, out, in_sizes[1]);
    angle_kernel<<<B_DIM, TPB, 0, stream>>>(target_angle, pred_angle, out);
}